// CentralityPositionalEncoding_56573309223853
// MI455X (gfx1250) — compile-verified
//
#include <hip/hip_runtime.h>
#include <hip/hip_bf16.h>
#include <stdint.h>

#define BLK 256

static constexpr int   NN     = 100000;   // N_NODES (reference constant)
static constexpr float DAMP_F = 0.85f;
static constexpr int   MAXIT  = 100;
static constexpr float TOL_F  = 1e-6f;
static constexpr float EPS_F  = 1e-8f;
// (1 - 0.85)/N computed in double like the reference, then narrowed
static constexpr float BASE_F = (float)((1.0 - 0.85) / (double)NN);

// ---- gfx1250 async global->LDS path (probe via __has_builtin) -------------
#if defined(__HIP_DEVICE_COMPILE__) && defined(__has_builtin)
#if __has_builtin(__builtin_amdgcn_global_load_async_to_lds_b128)
#define HAVE_ASYNC_LDS 1
#endif
#if __has_builtin(__builtin_amdgcn_wmma_f32_16x16x4_f32)
#define HAVE_WMMA_F32K4 1
#endif
#endif
#ifndef HAVE_ASYNC_LDS
#define HAVE_ASYNC_LDS 0
#endif
#ifndef HAVE_WMMA_F32K4
#define HAVE_WMMA_F32K4 0
#endif

#if HAVE_ASYNC_LDS
// builtin signature (from clang diagnostic): first param is
// 'int __vector_size__(16) __device__ *' (addrspace(1)); LDS side is AS3.
typedef int v4i_t __attribute__((vector_size(16)));
typedef __attribute__((address_space(1))) v4i_t* gv4p_t;
typedef __attribute__((address_space(3))) v4i_t* lv4p_t;
#endif

#if HAVE_WMMA_F32K4
typedef float v2f_t __attribute__((ext_vector_type(2)));
typedef float v8f_t __attribute__((ext_vector_type(8)));
#endif

// ---- exact fp32 wave sum: V_WMMA_F32_16X16X4_F32 against all-ones B -------
// A(16x4): lane L has A[L%16][lane<16?0:2] in VGPR0 (=v), VGPR1 (=0).
// D[m][n] = diff_m + diff_{m+16}; lane n holds rows 0..7, lane n+16 rows 8..15.
__device__ __forceinline__ float wave_sum32(float v) {
#if HAVE_WMMA_F32K4
  v2f_t a; a[0] = v;    a[1] = 0.0f;
  v2f_t b; b[0] = 1.0f; b[1] = 1.0f;
  v8f_t c = {};
  c = __builtin_amdgcn_wmma_f32_16x16x4_f32(false, a, false, b, (short)0, c,
                                            false, false);
  float s = ((c[0] + c[1]) + (c[2] + c[3])) + ((c[4] + c[5]) + (c[6] + c[7]));
  s += __shfl_xor(s, 16, 32);   // rows 0..7 half + rows 8..15 half
  return s;
#else
  for (int off = 16; off > 0; off >>= 1) v += __shfl_xor(v, off, 32);
  return v;
#endif
}

// deterministic block sum (valid on thread 0); requires uniform control flow
__device__ __forceinline__ float block_sum256(float v) {
  __shared__ float red[BLK / 32];
  float ws = wave_sum32(v);
  int lane = threadIdx.x & 31;
  int wave = threadIdx.x >> 5;
  if (lane == 0) red[wave] = ws;
  __syncthreads();
  float total = 0.0f;
  if (threadIdx.x == 0) {
    int nw = (int)(blockDim.x >> 5);
    for (int i = 0; i < nw; ++i) total += red[i];   // fixed order
  }
  return total;
}

// ---------------------------------------------------------------------------
__global__ void k_init(float* __restrict__ out_deg, float* __restrict__ deg,
                       float* __restrict__ accum, float* __restrict__ pr,
                       float* __restrict__ blocksums, int nblk,
                       int* __restrict__ flags, int n) {
  int i = blockIdx.x * blockDim.x + threadIdx.x;
  if (i < n) {
    out_deg[i] = 0.0f;
    deg[i]     = 0.0f;
    accum[i]   = 0.0f;
    pr[i]      = 1.0f / (float)n;
  }
  if (i < nblk) blocksums[i] = 0.0f;
  if (i == 0) { flags[0] = 0; flags[1] = 0; }   // done, apply
}

// degree (unweighted) and out_degree (weighted) via segment-sum atomics
__global__ void k_degree(const int* __restrict__ row, const float* __restrict__ w,
                         float* __restrict__ out_deg, float* __restrict__ deg, int E) {
  int e0 = (blockIdx.x * blockDim.x + threadIdx.x) * 4;
  if (e0 + 3 < E) {
    int4   r  = *(const int4*)(row + e0);
    float4 ww = *(const float4*)(w + e0);
    atomicAdd(&out_deg[r.x], ww.x); atomicAdd(&deg[r.x], 1.0f);
    atomicAdd(&out_deg[r.y], ww.y); atomicAdd(&deg[r.y], 1.0f);
    atomicAdd(&out_deg[r.z], ww.z); atomicAdd(&deg[r.z], 1.0f);
    atomicAdd(&out_deg[r.w], ww.w); atomicAdd(&deg[r.w], 1.0f);
  } else {
    for (int k = 0; k < 4; ++k) {
      int e = e0 + k;
      if (e < E) {
        int r = row[e];
        atomicAdd(&out_deg[r], w[e]);
        atomicAdd(&deg[r], 1.0f);
      }
    }
  }
}

__global__ void k_invdeg(const int* __restrict__ row, const float* __restrict__ w,
                         const float* __restrict__ out_deg,
                         float* __restrict__ invd, int E) {
  int e0 = (blockIdx.x * blockDim.x + threadIdx.x) * 4;
  if (e0 + 3 < E) {
    int4   r  = *(const int4*)(row + e0);
    float4 ww = *(const float4*)(w + e0);
    float4 o;
    o.x = ww.x / (out_deg[r.x] + EPS_F);
    o.y = ww.y / (out_deg[r.y] + EPS_F);
    o.z = ww.z / (out_deg[r.z] + EPS_F);
    o.w = ww.w / (out_deg[r.w] + EPS_F);
    *(float4*)(invd + e0) = o;
  } else {
    for (int k = 0; k < 4; ++k) {
      int e = e0 + k;
      if (e < E) invd[e] = w[e] / (out_deg[row[e]] + EPS_F);
    }
  }
}

// contrib scatter: accum[col] += pr[row] * inv_deg   (skipped once converged)
__global__ void k_scatter(const int* __restrict__ row, const int* __restrict__ col,
                          const float* __restrict__ invd, const float* __restrict__ pr,
                          float* __restrict__ accum, const int* __restrict__ flags, int E) {
  if (flags[0]) return;   // done
  int tid = threadIdx.x;
  int e0  = (blockIdx.x * blockDim.x + tid) * 4;

#if HAVE_ASYNC_LDS
  __shared__ float s_inv[BLK * 4];
  bool staged = (e0 + 3 < E);
  if (staged) {
    // async stage this lane's 16B of inv_deg into LDS; overlaps with gathers
    __builtin_amdgcn_global_load_async_to_lds_b128(
        (gv4p_t)(invd + e0), (lv4p_t)&s_inv[tid * 4], 0, 0);
  }
#endif

  if (e0 + 3 < E) {
    int4 r = *(const int4*)(row + e0);
    int4 c = *(const int4*)(col + e0);
    float px = pr[r.x], py = pr[r.y], pz = pr[r.z], pw = pr[r.w];
#if HAVE_ASYNC_LDS
    asm volatile("s_wait_asynccnt 0" ::: "memory");
    float4 g = *(const float4*)(&s_inv[tid * 4]);
#else
    float4 g = *(const float4*)(invd + e0);
#endif
    atomicAdd(&accum[c.x], px * g.x);
    atomicAdd(&accum[c.y], py * g.y);
    atomicAdd(&accum[c.z], pz * g.z);
    atomicAdd(&accum[c.w], pw * g.w);
  } else {
    for (int k = 0; k < 4; ++k) {
      int e = e0 + k;
      if (e < E) atomicAdd(&accum[col[e]], pr[row[e]] * invd[e]);
    }
  }
}

// new_pr = base + damp*accum ; per-block L1-diff sums (WMMA) ; zero accum
__global__ void k_update(const float* __restrict__ pr, float* __restrict__ newpr,
                         float* __restrict__ accum, float* __restrict__ blocksums,
                         const int* __restrict__ flags, int n) {
  if (flags[0]) return;   // uniform across grid
  int i = blockIdx.x * blockDim.x + threadIdx.x;
  float diff = 0.0f;
  if (i < n) {
    float a = accum[i];
    accum[i] = 0.0f;
    float np = BASE_F + DAMP_F * a;
    newpr[i] = np;
    diff = fabsf(np - pr[i]);
  }
  // EXEC is all-ones here (guard above only masked the value)
  float total = block_sum256(diff);
  if (threadIdx.x == 0) blocksums[blockIdx.x] = total;
}

// deterministic fixed-order final reduce; latch done/apply flags
__global__ void k_flag(const float* __restrict__ blocksums, int nblk,
                       int* __restrict__ flags) {
  float s = 0.0f;
  for (int i = threadIdx.x; i < nblk; i += BLK) s += blocksums[i];
  float total = block_sum256(s);
  if (threadIdx.x == 0) {
    int  d    = flags[0];
    int  conv = (total < TOL_F) ? 1 : 0;
    int  nd   = (d || conv) ? 1 : 0;
    flags[0] = nd;            // done
    flags[1] = nd ? 0 : 1;    // apply new_pr this iteration?
  }
}

__global__ void k_commit(float* __restrict__ pr, const float* __restrict__ newpr,
                         const int* __restrict__ flags, int n) {
  if (!flags[1]) return;
  int i = blockIdx.x * blockDim.x + threadIdx.x;
  if (i < n) pr[i] = newpr[i];
}

// pos_enc[n,16] = [degree, pagerank, 0 ... 0]
__global__ void k_output(const float* __restrict__ deg, const float* __restrict__ pr,
                         float* __restrict__ out, int n) {
  int i = blockIdx.x * blockDim.x + threadIdx.x;
  if (i < n) {
    float4 z = make_float4(0.0f, 0.0f, 0.0f, 0.0f);
    float4 a = make_float4(deg[i], pr[i], 0.0f, 0.0f);
    float4* o = (float4*)(out + (size_t)i * 16);
    o[0] = a; o[1] = z; o[2] = z; o[3] = z;
  }
}

// ---------------------------------------------------------------------------
extern "C" void kernel_launch(void* const* d_in, const int* in_sizes, int n_in,
                              void* d_out, int out_size, void* d_ws, size_t ws_size,
                              hipStream_t stream) {
  const int E = in_sizes[2];          // edge_weight element count
  const int N = NN;                   // reference constant (scalar lives on device)

  const int*   row = (const int*)d_in[0];
  const int*   col = row + E;
  const float* w   = (const float*)d_in[2];
  float*       out = (float*)d_out;

  // workspace layout (256B-aligned slabs)
  char*  ws  = (char*)d_ws;
  size_t off = 0;
  auto carve = [&](size_t bytes) {
    char* p = ws + off;
    off = (off + bytes + 255) & ~(size_t)255;
    return p;
  };
  const int gridN  = (N + BLK - 1) / BLK;
  const int gridE4 = (E + 4 * BLK - 1) / (4 * BLK);

  float* out_deg   = (float*)carve((size_t)N * sizeof(float));
  float* deg       = (float*)carve((size_t)N * sizeof(float));
  float* pr        = (float*)carve((size_t)N * sizeof(float));
  float* newpr     = (float*)carve((size_t)N * sizeof(float));
  float* accum     = (float*)carve((size_t)N * sizeof(float));
  float* invd      = (float*)carve((size_t)E * sizeof(float));
  float* blocksums = (float*)carve((size_t)gridN * sizeof(float));
  int*   flags     = (int*)carve(2 * sizeof(int));
  (void)ws_size; (void)n_in; (void)out_size;

  k_init<<<gridN, BLK, 0, stream>>>(out_deg, deg, accum, pr, blocksums, gridN, flags, N);
  k_degree<<<gridE4, BLK, 0, stream>>>(row, w, out_deg, deg, E);
  k_invdeg<<<gridE4, BLK, 0, stream>>>(row, w, out_deg, invd, E);

  for (int it = 0; it < MAXIT; ++it) {
    k_scatter<<<gridE4, BLK, 0, stream>>>(row, col, invd, pr, accum, flags, E);
    k_update<<<gridN, BLK, 0, stream>>>(pr, newpr, accum, blocksums, flags, N);
    k_flag<<<1, BLK, 0, stream>>>(blocksums, gridN, flags);
    k_commit<<<gridN, BLK, 0, stream>>>(pr, newpr, flags, N);
  }

  k_output<<<gridN, BLK, 0, stream>>>(deg, pr, out, N);
}